// GCNBranch_4303557231017
// MI455X (gfx1250) — compile-verified
//
#include <hip/hip_runtime.h>
#include <stdint.h>

typedef __attribute__((ext_vector_type(2))) float v2f;
typedef __attribute__((ext_vector_type(8))) float v8f;

#define HIDDEN 128
#define APAD 4          // LDS row pad: stride 132 -> bank = (m*132+k)%64 varies with m

// ---------------------------------------------------------------------------
// zero a float region (grid-stride)
// ---------------------------------------------------------------------------
__global__ void gcn_zero_f32(float* __restrict__ p, long n) {
  long i = (long)blockIdx.x * blockDim.x + threadIdx.x;
  long stride = (long)gridDim.x * blockDim.x;
  for (; i < n; i += stride) p[i] = 0.0f;
}

// ---------------------------------------------------------------------------
// per-edge degree counting (u32 atomics -> global_atomic_add_u32)
// ---------------------------------------------------------------------------
__global__ void gcn_degree(const long long* __restrict__ src,
                           const long long* __restrict__ dst,
                           unsigned* __restrict__ deg_out,
                           unsigned* __restrict__ deg_in, int E) {
  int e = blockIdx.x * blockDim.x + threadIdx.x;
  if (e < E) {
    atomicAdd(&deg_out[(int)src[e]], 1u);
    atomicAdd(&deg_in[(int)dst[e]], 1u);
  }
}

// ---------------------------------------------------------------------------
// norms: rsqrt(max(deg,1))
// ---------------------------------------------------------------------------
__global__ void gcn_norms(const unsigned* __restrict__ deg_out,
                          const unsigned* __restrict__ deg_in,
                          float* __restrict__ norm_src,
                          float* __restrict__ norm_dst, int N) {
  int i = blockIdx.x * blockDim.x + threadIdx.x;
  if (i < N) {
    norm_src[i] = rsqrtf(fmaxf((float)deg_out[i], 1.0f));
    norm_dst[i] = rsqrtf(fmaxf((float)deg_in[i], 1.0f));
  }
}

// ---------------------------------------------------------------------------
// Pack weights into WMMA-B-fragment order:
//   Wp[(k/2)*OUT_DIM*2 + col*2 + 0] = W[k  ][col]
//   Wp[(k/2)*OUT_DIM*2 + col*2 + 1] = W[k+1][col]
// so each lane's B fragment (K pair for its column) is one contiguous float2.
// ---------------------------------------------------------------------------
__global__ void gcn_pack_w(const float* __restrict__ W, float* __restrict__ Wp,
                           int out_dim) {
  int i = blockIdx.x * blockDim.x + threadIdx.x;      // over (HIDDEN/2)*out_dim
  int total = (HIDDEN / 2) * out_dim;
  if (i < total) {
    int kp  = i / out_dim;       // K pair index
    int col = i - kp * out_dim;
    Wp[(long)i * 2 + 0] = W[(long)(2 * kp + 0) * out_dim + col];
    Wp[(long)i * 2 + 1] = W[(long)(2 * kp + 1) * out_dim + col];
  }
}

// ---------------------------------------------------------------------------
// SpMM scatter: agg[dst] += x[src] * (scale ? scale[src] : 1)
// one wave32 per edge, float4 per lane (128 floats), hw f32 atomics
// ---------------------------------------------------------------------------
__global__ __launch_bounds__(256)
void gcn_spmm(const float* __restrict__ x,
              const float* __restrict__ scale,   // norm_src or nullptr
              const long long* __restrict__ src,
              const long long* __restrict__ dst,
              float* __restrict__ agg, int E) {
  int e    = blockIdx.x * 8 + (threadIdx.x >> 5);   // 8 waves / block
  int lane = threadIdx.x & 31;
  if (e >= E) return;
  int s = (int)src[e];
  int d = (int)dst[e];
  float4 v = ((const float4*)(x + (long)s * HIDDEN))[lane];
  if (scale) {
    float sc = scale[s];
    v.x *= sc; v.y *= sc; v.z *= sc; v.w *= sc;
  }
  float* out = agg + (long)d * HIDDEN + lane * 4;
  unsafeAtomicAdd(out + 0, v.x);   // global_atomic_add_f32 (no return)
  unsafeAtomicAdd(out + 1, v.y);
  unsafeAtomicAdd(out + 2, v.z);
  unsafeAtomicAdd(out + 3, v.w);
}

// ---------------------------------------------------------------------------
// Fused GEMM stage:
//   out = [relu]( (agg * norm_dst) @ W + bias ) [* post]
// Block = 16 nodes; wave w computes output columns [16w,16w+16) with
// 32 chained V_WMMA_F32_16X16X4_F32 over K=128. A-tile staged in LDS.
// A frag (16x4 f32): lane(M=lane&15), vgpr pair = K {0,1}|{2,3} by lane-half.
// B frag (4x16 f32): lane(N=lane&15), vgpr pair = K rows -> packed float2.
// C/D (16x16 f32):   vgpr r holds rows r and r+8 across lane halves.
// ---------------------------------------------------------------------------
template<int OUT_DIM, bool RELU, bool POST>
__global__ __launch_bounds__(32 * (OUT_DIM / 16))
void gcn_gemm(const float* __restrict__ agg,
              const float* __restrict__ norm_dst,
              const float* __restrict__ Wp,     // packed [HIDDEN/2][OUT_DIM][2]
              const float* __restrict__ bias,   // [OUT_DIM]
              const float* __restrict__ post,   // norm_src or nullptr
              float* __restrict__ out, int N) {
  constexpr int WAVES = OUT_DIM / 16;
  __shared__ float Atile[16 * (HIDDEN + APAD)];

  const int row0 = blockIdx.x * 16;
  const int tid  = threadIdx.x;

  // cooperative load of 16x128 A-tile, pre-scaled by norm_dst
  for (int i = tid; i < 16 * HIDDEN; i += 32 * WAVES) {
    int m = i >> 7;          // /128
    int k = i & (HIDDEN - 1);
    int node = row0 + m;
    float v = 0.0f;
    if (node < N) v = agg[(long)node * HIDDEN + k] * norm_dst[node];
    Atile[m * (HIDDEN + APAD) + k] = v;
  }
  __syncthreads();

  const int lane = tid & 31;
  const int half = lane >> 4;      // 0: lanes 0-15, 1: lanes 16-31
  const int n16  = lane & 15;
  const int col  = (tid >> 5) * 16 + n16;

  const float* Arow = &Atile[n16 * (HIDDEN + APAD)];
  const float2* Bp  = (const float2*)Wp;   // [HIDDEN/2][OUT_DIM]
  v8f c = {};
#pragma unroll
  for (int k = 0; k < HIDDEN; k += 4) {
    const int kk = k + 2 * half;           // even -> kk/2 = k/2 + half
    v2f a, b;
    a.x = Arow[kk];
    a.y = Arow[kk + 1];
    float2 bf = Bp[(long)(k / 2 + half) * OUT_DIM + col];  // global_load_b64
    b.x = bf.x;
    b.y = bf.y;
    c = __builtin_amdgcn_wmma_f32_16x16x4_f32(
        /*neg_a=*/false, a, /*neg_b=*/false, b,
        /*c_mod=*/(short)0, c, /*reuse_a=*/false, /*reuse_b=*/false);
  }

  const float bv = bias[col];
#pragma unroll
  for (int r = 0; r < 8; ++r) {
    int node = row0 + r + 8 * half;
    if (node < N) {
      float v = c[r] + bv;
      if (RELU) v = fmaxf(v, 0.0f);
      if (POST) v *= post[node];
      out[(long)node * OUT_DIM + col] = v;
    }
  }
}

// ---------------------------------------------------------------------------
// inputs: feature[N*128] f32, src[E] i64, dst[E] i64,
//         W1[128*128] f32, b1[128] f32, W2[128*64] f32, b2[64] f32
// output: [N*64] f32
// ws: deg_out[N] deg_in[N] norm_src[N] norm_dst[N] Wp1[128*128] Wp2[128*64]
//     h1[N*128] agg[N*128]
// ---------------------------------------------------------------------------
extern "C" void kernel_launch(void* const* d_in, const int* in_sizes, int n_in,
                              void* d_out, int out_size, void* d_ws, size_t ws_size,
                              hipStream_t stream) {
  const float*     feature = (const float*)d_in[0];
  const long long* src     = (const long long*)d_in[1];
  const long long* dst     = (const long long*)d_in[2];
  const float*     W1      = (const float*)d_in[3];
  const float*     b1      = (const float*)d_in[4];
  const float*     W2      = (const float*)d_in[5];
  const float*     b2      = (const float*)d_in[6];
  float*           out     = (float*)d_out;

  const int N = in_sizes[0] / HIDDEN;   // 100000
  const int E = in_sizes[1];            // 1600000

  float* ws       = (float*)d_ws;
  float* deg_out  = ws;                       // N (as u32)
  float* deg_in   = ws + (long)N;             // N (as u32)
  float* norm_src = ws + 2L * N;
  float* norm_dst = ws + 3L * N;
  float* Wp1      = ws + 4L * N;              // 128*128
  float* Wp2      = Wp1 + HIDDEN * HIDDEN;    // 128*64
  float* h1       = Wp2 + HIDDEN * 64;        // N*128
  float* agg      = h1 + (long)N * HIDDEN;    // N*128

  const long aggN = (long)N * HIDDEN;
  const int  spmm_blocks = (E + 7) / 8;       // 8 edges (waves) per block

  // degrees + norms + weight packing
  gcn_zero_f32<<<512, 256, 0, stream>>>(deg_out, 2L * N);
  gcn_degree<<<(E + 255) / 256, 256, 0, stream>>>(src, dst,
      (unsigned*)deg_out, (unsigned*)deg_in, E);
  gcn_norms<<<(N + 255) / 256, 256, 0, stream>>>(
      (const unsigned*)deg_out, (const unsigned*)deg_in, norm_src, norm_dst, N);
  gcn_pack_w<<<(HIDDEN / 2 * HIDDEN + 255) / 256, 256, 0, stream>>>(W1, Wp1, HIDDEN);
  gcn_pack_w<<<(HIDDEN / 2 * 64 + 255) / 256, 256, 0, stream>>>(W2, Wp2, 64);

  // layer 1: agg = segsum_dst( feature[src] * norm_src[src] )
  gcn_zero_f32<<<4096, 256, 0, stream>>>(agg, aggN);
  gcn_spmm<<<spmm_blocks, 256, 0, stream>>>(feature, norm_src, src, dst, agg, E);
  // h1 = relu((agg*norm_dst)@W1 + b1) * norm_src   (pre-scale for layer 2)
  gcn_gemm<128, true, true><<<(N + 15) / 16, 256, 0, stream>>>(
      agg, norm_dst, Wp1, b1, norm_src, h1, N);

  // layer 2: agg = segsum_dst( h1[src] )  (already scaled)
  gcn_zero_f32<<<4096, 256, 0, stream>>>(agg, aggN);
  gcn_spmm<<<spmm_blocks, 256, 0, stream>>>(h1, nullptr, src, dst, agg, E);
  // out = (agg*norm_dst)@W2 + b2
  gcn_gemm<64, false, false><<<(N + 15) / 16, 128, 0, stream>>>(
      agg, norm_dst, Wp2, b2, nullptr, out, N);
}